// DualGraphLightGCN_Align_67568425501085
// MI455X (gfx1250) — compile-verified
//
#include <hip/hip_runtime.h>
#include <hip/hip_bf16.h>
#include <stdint.h>

#define DIM 64
#define BLOCK 256
#define WAVES_PER_BLOCK (BLOCK / 32)
#define EDGES_PER_WAVE 32
#define EDGES_PER_BLOCK (WAVES_PER_BLOCK * EDGES_PER_WAVE) // 256

static __host__ __device__ inline int cdiv(int a, int b) { return (a + b - 1) / b; }

// ---- CDNA5 async global->LDS staging (tracked by ASYNCcnt) -----------------
__device__ __forceinline__ void async_ld_b32(uint32_t lds_byte_off, const void* gaddr) {
  // gfx1250: LDS dest byte-address in a VGPR, 64-bit global VA in a VGPR pair.
  asm volatile("global_load_async_to_lds_b32 %0, %1, off"
               :: "v"(lds_byte_off), "v"(gaddr)
               : "memory");
}

__device__ __forceinline__ void wait_async0() {
#if __has_builtin(__builtin_amdgcn_s_wait_asynccnt)
  __builtin_amdgcn_s_wait_asynccnt(0);
#else
  asm volatile("s_wait_asynccnt 0" ::: "memory");
#endif
}

// ---- SpMM: out[dst] += val * x[src], COO, fp32 atomics ---------------------
// One wave owns 32 edges: stage (src,dst,val) triples into per-wave LDS with
// async loads, wait ASYNCcnt==0, then process 2 edges/iter with 16 lanes per
// edge, each lane doing a float4 gather + 4 global_atomic_add_f32 scatters.
__global__ __launch_bounds__(BLOCK) void spmm_atomic(
    const int* __restrict__ src, const int* __restrict__ dst,
    const float* __restrict__ val, const float* __restrict__ x,
    float* __restrict__ y, int nnz) {
  // Per wave: 32 src ints | 32 dst ints | 32 val words  = 96 dwords (384 B)
  __shared__ int smem[WAVES_PER_BLOCK * 96];
  const int wave = threadIdx.x >> 5;
  const int lane = threadIdx.x & 31;
  int* wbase = smem + wave * 96;
  const int waveBase = (blockIdx.x * WAVES_PER_BLOCK + wave) * EDGES_PER_WAVE;

  // Stage this wave's 32 edge triples into LDS (clamped at the tail so every
  // lane issues an in-bounds load; tail slots are simply never consumed).
  int e = waveBase + lane;
  int ec = (e < nnz) ? e : (nnz - 1);
  // Low 32 bits of a generic LDS pointer == workgroup-relative LDS byte addr.
  uint32_t lds0 = (uint32_t)(uintptr_t)(wbase + lane);
  async_ld_b32(lds0,       (const void*)(src + ec));
  async_ld_b32(lds0 + 128, (const void*)(dst + ec));
  async_ld_b32(lds0 + 256, (const void*)(val + ec));
  wait_async0();

  const int half = lane >> 4;        // which of 2 edges this lane serves
  const int comp = (lane & 15) * 4;  // float4 chunk within the 64-wide row

  #pragma unroll 4
  for (int i = 0; i < EDGES_PER_WAVE; i += 2) {
    const int slot = i + half;
    const int ge = waveBase + slot;
    if (ge < nnz) {
      const int   s = wbase[slot];                      // ds_load_b32 broadcast
      const int   d = wbase[32 + slot];
      const float v = __int_as_float(wbase[64 + slot]);
      const float4 xa = *reinterpret_cast<const float4*>(x + (size_t)s * DIM + comp);
      float* yp = y + (size_t)d * DIM + comp;
      atomicAdd(yp + 0, v * xa.x);
      atomicAdd(yp + 1, v * xa.y);
      atomicAdd(yp + 2, v * xa.z);
      atomicAdd(yp + 3, v * xa.w);
    }
  }
}

// ---- elementwise helpers (float4-vectorized) -------------------------------
__global__ void init_ui_kernel(const float4* __restrict__ uemb,
                               const float4* __restrict__ iemb,
                               float4* __restrict__ acc, float4* __restrict__ cur,
                               float4* __restrict__ nxt, int nUser4, int nTotal4) {
  int i = blockIdx.x * blockDim.x + threadIdx.x;
  if (i < nTotal4) {
    float4 v = (i < nUser4) ? uemb[i] : iemb[i - nUser4];
    acc[i] = v;
    cur[i] = v;
    nxt[i] = make_float4(0.f, 0.f, 0.f, 0.f);
  }
}

__global__ void init_ii_kernel(const float4* __restrict__ iemb,
                               float4* __restrict__ acc, float4* __restrict__ cur,
                               float4* __restrict__ nxt, int n4) {
  int i = blockIdx.x * blockDim.x + threadIdx.x;
  if (i < n4) {
    float4 v = iemb[i];
    acc[i] = v;
    cur[i] = v;
    nxt[i] = make_float4(0.f, 0.f, 0.f, 0.f);
  }
}

// acc += nxt; cur = nxt; nxt = 0   (prepares the next SpMM layer)
__global__ void post_layer_kernel(float4* __restrict__ acc, float4* __restrict__ cur,
                                  float4* __restrict__ nxt, int n4) {
  int i = blockIdx.x * blockDim.x + threadIdx.x;
  if (i < n4) {
    float4 t = nxt[i];
    float4 a = acc[i];
    a.x += t.x; a.y += t.y; a.z += t.z; a.w += t.w;
    acc[i] = a;
    cur[i] = t;
    nxt[i] = make_float4(0.f, 0.f, 0.f, 0.f);
  }
}

// users: out = acc/4 ; items: out = 0.5*(acc/4) + 0.5*(accI/3)
__global__ void combine_kernel(const float4* __restrict__ acc,
                               const float4* __restrict__ accI,
                               float4* __restrict__ out, int nUser4, int nTotal4) {
  int i = blockIdx.x * blockDim.x + threadIdx.x;
  if (i < nTotal4) {
    float4 a = acc[i];
    if (i < nUser4) {
      out[i] = make_float4(a.x * 0.25f, a.y * 0.25f, a.z * 0.25f, a.w * 0.25f);
    } else {
      float4 b = accI[i - nUser4];
      const float cu = 0.5f * 0.25f;          // (1-SEM_W) / (N_LAYERS+1)
      const float cs = 0.5f * (1.0f / 3.0f);  // SEM_W / (SEM_LAYERS+1)
      out[i] = make_float4(cu * a.x + cs * b.x, cu * a.y + cs * b.y,
                           cu * a.z + cs * b.z, cu * a.w + cs * b.w);
    }
  }
}

extern "C" void kernel_launch(void* const* d_in, const int* in_sizes, int n_in,
                              void* d_out, int out_size, void* d_ws, size_t ws_size,
                              hipStream_t stream) {
  const float* user_emb = (const float*)d_in[0];
  const float* item_emb = (const float*)d_in[1];
  const float* ui_val   = (const float*)d_in[2];
  const float* ii_val   = (const float*)d_in[3];
  const int*   ui_src   = (const int*)d_in[4];
  const int*   ui_dst   = (const int*)d_in[5];
  const int*   ii_src   = (const int*)d_in[6];
  const int*   ii_dst   = (const int*)d_in[7];

  const int nUsers = in_sizes[0] / DIM;
  const int nItems = in_sizes[1] / DIM;
  const int nNodes = nUsers + nItems;
  const int uiNnz  = in_sizes[2];
  const int iiNnz  = in_sizes[3];

  // Workspace carve-up (fp32): 3 node-sized + 3 item-sized buffers (~154 MB).
  float* acc  = (float*)d_ws;
  float* cur  = acc  + (size_t)nNodes * DIM;
  float* nxt  = cur  + (size_t)nNodes * DIM;
  float* accI = nxt  + (size_t)nNodes * DIM;
  float* curI = accI + (size_t)nItems * DIM;
  float* nxtI = curI + (size_t)nItems * DIM;

  const int nTotal4 = nNodes * DIM / 4;
  const int nUser4  = nUsers * DIM / 4;
  const int nItem4  = nItems * DIM / 4;

  init_ui_kernel<<<cdiv(nTotal4, BLOCK), BLOCK, 0, stream>>>(
      (const float4*)user_emb, (const float4*)item_emb,
      (float4*)acc, (float4*)cur, (float4*)nxt, nUser4, nTotal4);
  init_ii_kernel<<<cdiv(nItem4, BLOCK), BLOCK, 0, stream>>>(
      (const float4*)item_emb, (float4*)accI, (float4*)curI, (float4*)nxtI, nItem4);

  // 3 LightGCN layers over the user-item graph (all 150k nodes).
  for (int l = 0; l < 3; ++l) {
    spmm_atomic<<<cdiv(uiNnz, EDGES_PER_BLOCK), BLOCK, 0, stream>>>(
        ui_src, ui_dst, ui_val, cur, nxt, uiNnz);
    post_layer_kernel<<<cdiv(nTotal4, BLOCK), BLOCK, 0, stream>>>(
        (float4*)acc, (float4*)cur, (float4*)nxt, nTotal4);
  }

  // 2 semantic layers over the item-item graph (50k items).
  for (int l = 0; l < 2; ++l) {
    spmm_atomic<<<cdiv(iiNnz, EDGES_PER_BLOCK), BLOCK, 0, stream>>>(
        ii_src, ii_dst, ii_val, curI, nxtI, iiNnz);
    post_layer_kernel<<<cdiv(nItem4, BLOCK), BLOCK, 0, stream>>>(
        (float4*)accI, (float4*)curI, (float4*)nxtI, nItem4);
  }

  combine_kernel<<<cdiv(nTotal4, BLOCK), BLOCK, 0, stream>>>(
      (const float4*)acc, (const float4*)accI, (float4*)d_out, nUser4, nTotal4);
}